// RandLANet_37065567764909
// MI455X (gfx1250) — compile-verified
//
#include <hip/hip_runtime.h>
#include <math.h>

#define NPTS 30000
#define MQ   7500
#define KNB  16
#define DX   64
#define DREL 64
#define DF   128

typedef __attribute__((ext_vector_type(2))) float v2f;
typedef __attribute__((ext_vector_type(8))) float v8f;

// ---------------------------------------------------------------------------
// Kernel 1: brute-force KNN, one wave32 per query.
// Per-lane register-resident sorted top-16, then 32-way merge via LDS+shuffles.
// ---------------------------------------------------------------------------
__global__ __launch_bounds__(32) void knn_kernel(const float* __restrict__ pos,
                                                 const int* __restrict__ idx,
                                                 int* __restrict__ nbr) {
    const int m    = blockIdx.x;
    const int lane = threadIdx.x;

    const int q = idx[m];
    const float qx = pos[q * 3 + 0];
    const float qy = pos[q * 3 + 1];
    const float qz = pos[q * 3 + 2];

    const float FINF = 3.4e38f;
    float d[KNB];
    int   id[KNB];
#pragma unroll
    for (int i = 0; i < KNB; ++i) { d[i] = FINF; id[i] = -1; }

    for (int p = lane; p < NPTS; p += 32) {
        const float dx = qx - pos[p * 3 + 0];
        const float dy = qy - pos[p * 3 + 1];
        const float dz = qz - pos[p * 3 + 2];
        const float dd = dx * dx + dy * dy + dz * dz;
        if (dd < d[KNB - 1]) {
            d[KNB - 1]  = dd;
            id[KNB - 1] = p;
#pragma unroll
            for (int i = KNB - 1; i > 0; --i) {
                const bool c = d[i] < d[i - 1];
                const float ta = c ? d[i - 1] : d[i];
                const float tb = c ? d[i]     : d[i - 1];
                const int   ia = c ? id[i - 1] : id[i];
                const int   ib = c ? id[i]     : id[i - 1];
                d[i] = ta; d[i - 1] = tb;
                id[i] = ia; id[i - 1] = ib;
            }
        }
    }

    __shared__ float sd[32 * KNB];
    __shared__ int   si[32 * KNB];
#pragma unroll
    for (int i = 0; i < KNB; ++i) {
        sd[lane * KNB + i] = d[i];
        si[lane * KNB + i] = id[i];
    }
    __syncthreads();

    // 32-way merge of sorted lists: 16 rounds of cross-lane min selection.
    int p = 0;
    for (int r = 0; r < KNB; ++r) {
        float bd = (p < KNB) ? sd[lane * KNB + p] : FINF;
        int   bl = lane;
#pragma unroll
        for (int off = 16; off > 0; off >>= 1) {
            const float od = __shfl_xor(bd, off, 32);
            const int   ol = __shfl_xor(bl, off, 32);
            if (od < bd || (od == bd && ol < bl)) { bd = od; bl = ol; }
        }
        const int pw   = __shfl(p, bl, 32);
        const int widx = si[bl * KNB + pw];   // broadcast read
        if (lane == 0) nbr[m * KNB + r] = widx;
        if (lane == bl) ++p;
    }
}

// ---------------------------------------------------------------------------
// Kernel 2: fused LocSE + attentive pooling. One 256-thread block per query.
// g = fij @ W_att computed with v_wmma_f32_16x16x4_f32 (8 waves x 8 col-tiles).
// ---------------------------------------------------------------------------
__global__ __launch_bounds__(256) void encode_attn_kernel(
    const float* __restrict__ x, const float* __restrict__ pos,
    const int* __restrict__ idx, const int* __restrict__ nbr,
    const float* __restrict__ W_pos, const float* __restrict__ b_pos,
    const float* __restrict__ W_att, const float* __restrict__ b_att,
    float* __restrict__ aggr) {
    const int m = blockIdx.x;
    const int t = threadIdx.x;

    __shared__ float fij[KNB][DF];   // 8 KB
    __shared__ float g[KNB][DF];     // 8 KB
    __shared__ float rel[KNB][12];
    __shared__ int   nb[KNB];
    __shared__ float rowmax[KNB], rowsum[KNB];

    if (t < KNB) nb[t] = nbr[m * KNB + t];
    __syncthreads();

    if (t < KNB) {
        const int j = nb[t];
        const int q = idx[m];
        const float ix = pos[q * 3 + 0], iy = pos[q * 3 + 1], iz = pos[q * 3 + 2];
        const float jx = pos[j * 3 + 0], jy = pos[j * 3 + 1], jz = pos[j * 3 + 2];
        const float vx = ix - jx, vy = iy - jy, vz = iz - jz;
        const float dn = sqrtf(vx * vx + vy * vy + vz * vz);
        rel[t][0] = ix; rel[t][1] = iy; rel[t][2] = iz;
        rel[t][3] = jx; rel[t][4] = jy; rel[t][5] = jz;
        rel[t][6] = vx; rel[t][7] = vy; rel[t][8] = vz;
        rel[t][9] = dn;
    }
    __syncthreads();

    // fij[:, 0:64] = x[nbr]; fij[:, 64:128] = relu(rel @ W_pos + b_pos)
    for (int e = t; e < KNB * DX; e += 256) {
        const int r = e >> 6, c = e & 63;
        fij[r][c] = x[nb[r] * DX + c];
    }
    for (int e = t; e < KNB * DREL; e += 256) {
        const int r = e >> 6, c = e & 63;
        float acc = b_pos[c];
#pragma unroll
        for (int k = 0; k < 10; ++k) acc += rel[r][k] * W_pos[k * DREL + c];
        fij[r][DX + c] = fmaxf(acc, 0.0f);
    }
    __syncthreads();

    // --- WMMA: wave w computes column tile n0 = 16*w of g = fij @ W_att ---
    const int lane = t & 31;
    const int wv   = t >> 5;          // 0..7
    const int n0   = wv * 16;
    const int row  = lane & 15;       // M (A) / N (B,C,D)
    const int hi   = lane >> 4;       // lane half -> K offset 0 or 2
    const int col  = n0 + row;

    v8f c = {};
    for (int kb = 0; kb < DF; kb += 4) {
        v2f a, b;
        const int ka = kb + 2 * hi;
        a.x = fij[row][ka];
        a.y = fij[row][ka + 1];
        b.x = W_att[(ka)     * DF + col];
        b.y = W_att[(ka + 1) * DF + col];
        c = __builtin_amdgcn_wmma_f32_16x16x4_f32(false, a, false, b,
                                                  (short)0, c, false, false);
    }
#pragma unroll
    for (int r = 0; r < 8; ++r) {
        g[r + 8 * hi][col] = c[r] + b_att[col];   // C/D layout: VGPR r -> M = r + 8*hi
    }
    __syncthreads();

    // softmax stats over feature dim (axis=-1), one thread per neighbor row
    if (t < KNB) {
        float mx = -3.4e38f;
        for (int cc = 0; cc < DF; ++cc) mx = fmaxf(mx, g[t][cc]);
        float s = 0.0f;
        for (int cc = 0; cc < DF; ++cc) s += __expf(g[t][cc] - mx);
        rowmax[t] = mx; rowsum[t] = s;
    }
    __syncthreads();

    // aggr[c] = mean_r softmax(g)[r][c] * fij[r][c]
    if (t < DF) {
        float acc = 0.0f;
#pragma unroll
        for (int r = 0; r < KNB; ++r) {
            const float sv = __expf(g[r][t] - rowmax[r]) / rowsum[r];
            acc += sv * fij[r][t];
        }
        aggr[m * DF + t] = acc * (1.0f / (float)KNB);
    }
}

// ---------------------------------------------------------------------------
// Kernel 3: out = relu(aggr @ W_glob + b_glob), WMMA f32, 16-row tiles.
// ---------------------------------------------------------------------------
__global__ __launch_bounds__(256) void global_nn_kernel(
    const float* __restrict__ aggr, const float* __restrict__ W_glob,
    const float* __restrict__ b_glob, float* __restrict__ out) {
    const int t    = threadIdx.x;
    const int lane = t & 31;
    const int wv   = t >> 5;
    const int n0   = wv * 16;
    const int row  = lane & 15;
    const int hi   = lane >> 4;
    const int col  = n0 + row;
    const int r0   = blockIdx.x * 16;

    int arow = r0 + row;
    if (arow >= MQ) arow = MQ - 1;   // clamp reads; stores are guarded below

    v8f c = {};
    for (int kb = 0; kb < DF; kb += 4) {
        const int ka = kb + 2 * hi;
        v2f a, b;
        a.x = aggr[arow * DF + ka];
        a.y = aggr[arow * DF + ka + 1];
        b.x = W_glob[(ka)     * DF + col];
        b.y = W_glob[(ka + 1) * DF + col];
        c = __builtin_amdgcn_wmma_f32_16x16x4_f32(false, a, false, b,
                                                  (short)0, c, false, false);
    }
#pragma unroll
    for (int r = 0; r < 8; ++r) {
        const int orow = r0 + r + 8 * hi;
        if (orow < MQ)
            out[orow * DF + col] = fmaxf(c[r] + b_glob[col], 0.0f);
    }
}

// ---------------------------------------------------------------------------
extern "C" void kernel_launch(void* const* d_in, const int* in_sizes, int n_in,
                              void* d_out, int out_size, void* d_ws, size_t ws_size,
                              hipStream_t stream) {
    (void)in_sizes; (void)n_in; (void)out_size; (void)ws_size;

    const float* x      = (const float*)d_in[0];
    const float* pos    = (const float*)d_in[1];
    const int*   idx    = (const int*)  d_in[2];
    const float* W_pos  = (const float*)d_in[3];
    const float* b_pos  = (const float*)d_in[4];
    const float* W_att  = (const float*)d_in[5];
    const float* b_att  = (const float*)d_in[6];
    const float* W_glob = (const float*)d_in[7];
    const float* b_glob = (const float*)d_in[8];
    float*       out    = (float*)d_out;

    int*   nbr  = (int*)d_ws;
    size_t off  = ((size_t)MQ * KNB * sizeof(int) + 255) & ~(size_t)255;
    float* aggr = (float*)((char*)d_ws + off);

    knn_kernel<<<MQ, 32, 0, stream>>>(pos, idx, nbr);
    encode_attn_kernel<<<MQ, 256, 0, stream>>>(x, pos, idx, nbr,
                                               W_pos, b_pos, W_att, b_att, aggr);
    global_nn_kernel<<<(MQ + 15) / 16, 256, 0, stream>>>(aggr, W_glob, b_glob, out);
}